// HyenaFilter_62225486184536
// MI455X (gfx1250) — compile-verified
//
#include <hip/hip_runtime.h>
#include <stdint.h>

// ---------------------------------------------------------------------------
// Hyena filter + FFT long convolution for MI455X (gfx1250).
//   Kernel 1: implicit filter MLP with V_WMMA_F32_16X16X4_F32 (fp32 matrix pipe)
//   Kernel 2: per-channel causal conv via radix-4 Stockham FFT (N=16384) in LDS,
//             real-packing trick (FFT(x + i*k)), TDM async load of the filter row.
// ---------------------------------------------------------------------------

#define D_MODEL 1024
#define SEQ_L   8192
#define FFT_N   16384
#define ORDER   64

typedef float  v2f __attribute__((ext_vector_type(2)));
typedef float  v8f __attribute__((ext_vector_type(8)));
typedef unsigned int v4u __attribute__((ext_vector_type(4)));
typedef int    v8i __attribute__((ext_vector_type(8)));
typedef int    v4i __attribute__((ext_vector_type(4)));

#define TWO_PI 6.283185307179586f
// |log(0.01)/1.5| and |log(0.01)/0.3|
#define MIN_DECAY_ABS 3.0701134573253943f
#define MAX_DECAY_ABS 15.350567286626971f

// ============================ Kernel 1: filter MLP ==========================
//
// WMMA f32 16x16x4 operand layouts (CDNA5 ISA 7.12.2, wave32):
//   A (16x4):  lane l, vgpr v -> A[M = l&15][K = 4*ks + 2*(l>>4) + v]
//   B (4x16):  lane l, vgpr v -> B[K = 4*ks + 2*v + (l>>4)][N = l&15]  (assumed,
//              mirrored from the documented 8/16-bit B layouts)
//   C/D:       lane l, vgpr j -> C[M = j + 8*(l>>4)][N = l&15]

__device__ inline v8f wmma_mm16(const float* A, int lda,
                                const float* B, int ldb,
                                int nbase, int K, int lane) {
  v8f acc = {0.f, 0.f, 0.f, 0.f, 0.f, 0.f, 0.f, 0.f};
  const int m  = lane & 15;
  const int hi = lane >> 4;
  for (int ks = 0; ks < K; ks += 4) {
    v2f a, b;
    a.x = A[m * lda + ks + 2 * hi + 0];
    a.y = A[m * lda + ks + 2 * hi + 1];
    b.x = B[(ks + 0 + hi) * ldb + nbase + m];
    b.y = B[(ks + 2 + hi) * ldb + nbase + m];
    acc = __builtin_amdgcn_wmma_f32_16x16x4_f32(false, a, false, b,
                                                (short)0, acc, false, false);
  }
  return acc;
}

__device__ inline void store_sin_act(float* H, int nt,
                                     const float* __restrict__ bias,
                                     const float* __restrict__ freq,
                                     v8f acc, int lane) {
  const int m  = lane & 15;
  const int hi = lane >> 4;
  const int col = nt * 16 + m;
  const float bv = bias[col];
  const float fv = freq[col];
#pragma unroll
  for (int j = 0; j < 8; ++j) {
    const int row = j + 8 * hi;
    H[row * ORDER + col] = __sinf(fv * (acc[j] + bv));
  }
}

__global__ void __launch_bounds__(128)
hyena_filter_kernel(const float* __restrict__ lin_in_w,
                    const float* __restrict__ lin_in_b,
                    const float* __restrict__ freq,
                    const float* __restrict__ w1g, const float* __restrict__ b1g,
                    const float* __restrict__ w2g, const float* __restrict__ b2g,
                    const float* __restrict__ out_w,
                    float* __restrict__ kbuf) {
  __shared__ float zt[16 * 40];   // 16 rows x EMB(33) padded to K=36, stride 40
  __shared__ float w0[36 * 64];   // lin_in_w zero-padded to 36 rows
  __shared__ float hA[16 * ORDER];
  __shared__ float hB[16 * ORDER];

  const int tid  = threadIdx.x;
  const int lane = tid & 31;
  const int wave = tid >> 5;
  const int rowBase = blockIdx.x * 16;

  // --- stage positional-embedding tile z[16][36] ---
  for (int e = tid; e < 16 * 36; e += 128) {
    const int i = e / 36, c = e % 36;
    const int r = rowBase + i;
    float val = 0.f;
    if (c == 0) {
      val = (float)r * (1.0f / (float)(SEQ_L - 1));        // t
    } else if (c < 33) {
      const int band  = (c - 1) & 15;
      const bool isSin = (c >= 17);
      const float fb = 1e-4f + (float)band * ((15.0f - 1e-4f) / 15.0f);
      const float wj = TWO_PI * (float)r * (1.0f / (float)SEQ_L);
      float sv, cv;
      __sincosf(fb * wj, &sv, &cv);
      val = isSin ? -sv : cv;                               // z = [t, cos, -sin]
    }
    zt[i * 40 + c] = val;
  }
  // --- stage zero-padded first-layer weights ---
  for (int e = tid; e < 36 * 64; e += 128) {
    const int rr = e >> 6, cc = e & 63;
    w0[e] = (rr < 33) ? lin_in_w[rr * 64 + cc] : 0.f;
  }
  __syncthreads();

  // Layer 0: [16x36] @ [36x64]  (4 N-tiles, one per wave)
  { v8f acc = wmma_mm16(zt, 40, w0, 64, wave * 16, 36, lane);
    store_sin_act(hA, wave, lin_in_b, freq, acc, lane); }
  __syncthreads();
  // Layer 1: [16x64] @ [64x64]
  { v8f acc = wmma_mm16(hA, 64, w1g, 64, wave * 16, 64, lane);
    store_sin_act(hB, wave, b1g, freq, acc, lane); }
  __syncthreads();
  // Layer 2: [16x64] @ [64x64]
  { v8f acc = wmma_mm16(hB, 64, w2g, 64, wave * 16, 64, lane);
    store_sin_act(hA, wave, b2g, freq, acc, lane); }
  __syncthreads();

  // Output projection [16x64] @ [64x1024] + exponential modulation,
  // written transposed: kbuf[d][l].
  for (int nt = wave; nt < 64; nt += 4) {
    v8f acc = wmma_mm16(hA, 64, out_w, D_MODEL, nt * 16, 64, lane);
    const int m  = lane & 15;
    const int hi = lane >> 4;
    const int d  = nt * 16 + m;
    const float dabs = MIN_DECAY_ABS +
        (float)d * ((MAX_DECAY_ABS - MIN_DECAY_ABS) / (float)(D_MODEL - 1));
#pragma unroll
    for (int j = 0; j < 8; ++j) {
      const int r = rowBase + j + 8 * hi;
      const float t = (float)r * (1.0f / (float)(SEQ_L - 1));
      kbuf[(size_t)d * SEQ_L + r] = acc[j] * __expf(-t * dabs);
    }
  }
}

// ========================= Kernel 2: FFT convolution ========================

__device__ inline float2 cmul(float2 a, float2 b) {
  return make_float2(a.x * b.x - a.y * b.y, a.x * b.y + a.y * b.x);
}

// Radix-4 Stockham butterfly (auto-sorting).  jsgn = -1 forward, +1 inverse.
__device__ inline void bf4(const float2* __restrict__ src, float2* __restrict__ dst,
                           int q, int p, int s, int m, float ang1, float jsgn) {
  float2 a = src[q + s * p];
  float2 b = src[q + s * (p + m)];
  float2 c = src[q + s * (p + 2 * m)];
  float2 e = src[q + s * (p + 3 * m)];
  float2 apc = make_float2(a.x + c.x, a.y + c.y);
  float2 amc = make_float2(a.x - c.x, a.y - c.y);
  float2 bpe = make_float2(b.x + e.x, b.y + e.y);
  float2 bme = make_float2(b.x - e.x, b.y - e.y);
  float2 u0 = make_float2(apc.x + bpe.x, apc.y + bpe.y);
  float2 u2 = make_float2(apc.x - bpe.x, apc.y - bpe.y);
  float2 u1 = make_float2(amc.x - jsgn * bme.y, amc.y + jsgn * bme.x);
  float2 u3 = make_float2(amc.x + jsgn * bme.y, amc.y - jsgn * bme.x);
  float sn, cs;
  __sincosf(ang1, &sn, &cs);
  float2 w1 = make_float2(cs, sn);
  float2 w2 = cmul(w1, w1);
  float2 w3 = cmul(w2, w1);
  dst[q + s * (4 * p + 0)] = u0;
  dst[q + s * (4 * p + 1)] = cmul(w1, u1);
  dst[q + s * (4 * p + 2)] = cmul(w2, u2);
  dst[q + s * (4 * p + 3)] = cmul(w3, u3);
}

__global__ void __launch_bounds__(512)
hyena_fftconv_kernel(const float* __restrict__ x,
                     const float* __restrict__ kbuf,
                     const float* __restrict__ Dvec,
                     float* __restrict__ out) {
  extern __shared__ __align__(16) char smem_raw[];
  float2* bufA = (float2*)smem_raw;            // 16384 complex = 128 KB
  float2* bufB = bufA + FFT_N;                 // 16384 complex = 128 KB
  float*  kst  = (float*)(bufB + FFT_N);       // 8192 floats   =  32 KB (TDM dest)

  const int tid = threadIdx.x;
  const int bd  = blockIdx.x;                  // b*1024 + d
  const int d   = bd & (D_MODEL - 1);
  const float* xrow = x + (size_t)bd * SEQ_L;
  const float* krow = kbuf + (size_t)d * SEQ_L;

  // --- TDM: async DMA of the 32 KB filter row into LDS (wave 0 issues it) ---
  if ((tid >> 5) == 0) {
    const unsigned long long ga = (unsigned long long)(uintptr_t)krow;
    // D# group0: count=1 | lds_addr=262144 | global_addr | type=2
    v4u g0 = { 1u,
               (unsigned)(2u * FFT_N * sizeof(float2)),
               (unsigned)(ga & 0xFFFFFFFFu),
               (unsigned)((ga >> 32) & 0x1FFFFFFu) | (2u << 30) };
    // D# group1: data_size=4B, tensor_dim0=8192, tensor_dim1=1, tile_dim0=8192,
    //            tile_dim1=1, tensor_dim0_stride=8192
    v8i g1 = { (int)(2u << 16),
               (int)((8192u & 0xFFFFu) << 16),
               (int)((8192u >> 16) | (1u << 16)),
               (int)(8192u << 16),
               1,
               8192,
               0, 0 };
    v4i g2 = {0, 0, 0, 0};
    v4i g3 = {0, 0, 0, 0};
    v8i g4 = {0, 0, 0, 0, 0, 0, 0, 0};
    __builtin_amdgcn_tensor_load_to_lds(g0, g1, g2, g3, g4, 0);
    __builtin_amdgcn_s_wait_tensorcnt(0);
  }

  // z[n] = x[n] + i*k[n] (real-packing: one FFT covers both spectra), zero-pad.
  for (int i = tid; i < FFT_N; i += 512) {
    const float xv = (i < SEQ_L) ? xrow[i] : 0.f;
    bufA[i] = make_float2(xv, 0.f);
  }
  __syncthreads();
  for (int i = tid; i < SEQ_L; i += 512) bufA[i].y = kst[i];
  __syncthreads();

  float2* src = bufA;
  float2* dst = bufB;

  // ---- forward FFT: 7 radix-4 Stockham passes ----
  {
    int s = 1, log2s = 0, n = FFT_N;
    for (int pass = 0; pass < 7; ++pass) {
      const int m = n >> 2;
      const float ang_step = -TWO_PI / (float)n;
      for (int idx = tid; idx < FFT_N / 4; idx += 512) {
        const int q = idx & (s - 1);
        const int p = idx >> log2s;
        bf4(src, dst, q, p, s, m, ang_step * (float)p, -1.0f);
      }
      __syncthreads();
      float2* tmp = src; src = dst; dst = tmp;
      s <<= 2; log2s += 2; n >>= 2;
    }
  }

  // ---- Hermitian split + spectral multiply: W = X(f) * K(f) ----
  for (int f = tid; f < FFT_N; f += 512) {
    const float2 zf = src[f];
    const float2 zc = src[(FFT_N - f) & (FFT_N - 1)];
    const float Xr = 0.5f * (zf.x + zc.x);
    const float Xi = 0.5f * (zf.y - zc.y);
    const float Kr = 0.5f * (zf.y + zc.y);
    const float Ki = -0.5f * (zf.x - zc.x);
    dst[f] = make_float2(Xr * Kr - Xi * Ki, Xr * Ki + Xi * Kr);
  }
  __syncthreads();
  { float2* tmp = src; src = dst; dst = tmp; }

  // ---- inverse FFT (unscaled; 1/N folded into the store) ----
  {
    int s = 1, log2s = 0, n = FFT_N;
    for (int pass = 0; pass < 7; ++pass) {
      const int m = n >> 2;
      const float ang_step = TWO_PI / (float)n;
      for (int idx = tid; idx < FFT_N / 4; idx += 512) {
        const int q = idx & (s - 1);
        const int p = idx >> log2s;
        bf4(src, dst, q, p, s, m, ang_step * (float)p, 1.0f);
      }
      __syncthreads();
      float2* tmp = src; src = dst; dst = tmp;
      s <<= 2; log2s += 2; n >>= 2;
    }
  }

  // ---- causal slice + skip connection ----
  const float Dd = Dvec[d];
  const float invN = 1.0f / (float)FFT_N;
  for (int l = tid; l < SEQ_L; l += 512) {
    out[(size_t)bd * SEQ_L + l] = src[l].x * invN + xrow[l] * Dd;
  }
}

// ================================ Launch ====================================

extern "C" void kernel_launch(void* const* d_in, const int* in_sizes, int n_in,
                              void* d_out, int out_size, void* d_ws, size_t ws_size,
                              hipStream_t stream) {
  const float* x     = (const float*)d_in[0];
  // d_in[1] is L (constant 8192, baked in)
  const float* lin_w = (const float*)d_in[2];
  const float* lin_b = (const float*)d_in[3];
  const float* freq  = (const float*)d_in[4];
  const float* w1    = (const float*)d_in[5];
  const float* b1    = (const float*)d_in[6];
  const float* w2    = (const float*)d_in[7];
  const float* b2    = (const float*)d_in[8];
  const float* outw  = (const float*)d_in[9];
  const float* Dv    = (const float*)d_in[10];
  float* out  = (float*)d_out;
  float* kbuf = (float*)d_ws;   // 1024*8192 floats = 32 MB of workspace

  const int shmem = 2 * FFT_N * (int)sizeof(float2) + SEQ_L * (int)sizeof(float);
  (void)hipFuncSetAttribute(reinterpret_cast<const void*>(hyena_fftconv_kernel),
                            hipFuncAttributeMaxDynamicSharedMemorySize, shmem);

  hyena_filter_kernel<<<SEQ_L / 16, 128, 0, stream>>>(
      lin_w, lin_b, freq, w1, b1, w2, b2, outw, kbuf);
  hyena_fftconv_kernel<<<2 * D_MODEL, 512, shmem, stream>>>(x, kbuf, Dv, out);
}